// Net_13451837571225
// MI455X (gfx1250) — compile-verified
//
#include <hip/hip_runtime.h>
#include <hip/hip_bf16.h>

typedef float v2f __attribute__((ext_vector_type(2)));
typedef float v8f __attribute__((ext_vector_type(8)));

#define N_NODES   100000
#define IN_FEAT   512
#define HID       32

// ---------------------------------------------------------------------------
// Zero a float buffer
// ---------------------------------------------------------------------------
__global__ void zero_f32(float* __restrict__ p, int n) {
    int i = blockIdx.x * blockDim.x + threadIdx.x;
    if (i < n) p[i] = 0.0f;
}

// ---------------------------------------------------------------------------
// Degree count: one thread per edge, atomic add 1.0 to deg[dst]
// ---------------------------------------------------------------------------
__global__ void degree_kernel(const long long* __restrict__ dst,
                              float* __restrict__ deg, int E) {
    int e = blockIdx.x * blockDim.x + threadIdx.x;
    if (e < E) atomicAdd(&deg[(int)dst[e]], 1.0f);
}

// ---------------------------------------------------------------------------
// dinv[i] = rsqrt(deg[i] + 1)   (in place)
// ---------------------------------------------------------------------------
__global__ void rsqrt_kernel(float* __restrict__ deg, int n) {
    int i = blockIdx.x * blockDim.x + threadIdx.x;
    if (i < n) deg[i] = rsqrtf(deg[i] + 1.0f);
}

// ---------------------------------------------------------------------------
// WMMA GEMM:  H[nrows x 32] = (RELU? relu(A) : A)[nrows x K] @ W[K x 32]
// Uses V_WMMA_F32_16X16X4_F32 (full-fp32 matrix op, CDNA5).
// Block = 256 threads = 8 waves; each wave computes one 16-row tile with two
// 16x16 accumulators. W is staged in LDS ([K][32] row-major).
//
// A fragment (16x4 f32, 2 VGPRs/lane):
//   lane l: m = l&15, kh = l>>4 ; a.x = A[m][k+2*kh], a.y = A[m][k+2*kh+1]
// B fragment (4x16 f32, 2 VGPRs/lane), symmetric layout:
//   lane l: n = l&15 ; b.x = W[k+2*kh][n], b.y = W[k+2*kh+1][n]
// C/D (16x16 f32, 8 VGPRs): vgpr r -> row r + 8*kh, col n
// ---------------------------------------------------------------------------
template <int RELU>
__global__ __launch_bounds__(256) void gemm_wmma(const float* __restrict__ A,
                                                 const float* __restrict__ W,
                                                 float* __restrict__ H,
                                                 int ntiles, int K) {
    extern __shared__ float sW[];  // K * 32 floats

    // cooperative load of W into LDS
    for (int i = threadIdx.x; i < K * HID; i += blockDim.x) sW[i] = W[i];
    __syncthreads();

    const int wave = threadIdx.x >> 5;
    const int lane = threadIdx.x & 31;
    const int tile = blockIdx.x * 8 + wave;
    if (tile >= ntiles) return;

    const int m  = lane & 15;   // row within tile / output column
    const int kh = lane >> 4;   // which K-pair this half-wave holds

    const size_t row = (size_t)tile * 16 + m;
    const float* arow = A + row * (size_t)K + 2 * kh;

    v8f acc0 = {};  // cols 0..15
    v8f acc1 = {};  // cols 16..31

#pragma unroll 4
    for (int k = 0; k < K; k += 4) {
        v2f a;
        a.x = arow[k];
        a.y = arow[k + 1];
        if (RELU) {
            a.x = fmaxf(a.x, 0.0f);
            a.y = fmaxf(a.y, 0.0f);
        }
        const int kr = k + 2 * kh;
        v2f b0, b1;
        b0.x = sW[kr * HID + m];
        b0.y = sW[(kr + 1) * HID + m];
        b1.x = sW[kr * HID + 16 + m];
        b1.y = sW[(kr + 1) * HID + 16 + m];

        acc0 = __builtin_amdgcn_wmma_f32_16x16x4_f32(
            false, a, false, b0, (short)0, acc0, false, false);
        acc1 = __builtin_amdgcn_wmma_f32_16x16x4_f32(
            false, a, false, b1, (short)0, acc1, false, false);
    }

    // write back: vgpr r -> row tile*16 + r + 8*kh, col m (and m+16)
    const size_t base = ((size_t)tile * 16 + 8 * kh) * HID;
#pragma unroll
    for (int r = 0; r < 8; ++r) {
        H[base + (size_t)r * HID + m]      = acc0[r];
        H[base + (size_t)r * HID + m + 16] = acc1[r];
    }
}

// ---------------------------------------------------------------------------
// agg[i][f] = h[i][f] * dinv[i]^2 + b[f]      (self-loop term + bias)
// ---------------------------------------------------------------------------
__global__ void init_agg_kernel(const float* __restrict__ h,
                                const float* __restrict__ dinv,
                                const float* __restrict__ b,
                                float* __restrict__ agg, int n) {
    int idx = blockIdx.x * blockDim.x + threadIdx.x;
    if (idx >= n) return;
    int node = idx >> 5;   // HID == 32
    int f    = idx & 31;
    float di = dinv[node];
    agg[idx] = h[idx] * di * di + b[f];
}

// ---------------------------------------------------------------------------
// Edge scatter: one wave per edge, lane = feature.
//   agg[dst][lane] += h[src][lane] * dinv[src]*dinv[dst]
// Coalesced 128B gather + 128B global_atomic_add_f32 per edge.
// ---------------------------------------------------------------------------
__global__ __launch_bounds__(256) void scatter_kernel(const long long* __restrict__ src,
                                                      const long long* __restrict__ dst,
                                                      const float* __restrict__ dinv,
                                                      const float* __restrict__ h,
                                                      float* __restrict__ agg, int E) {
    int gtid = blockIdx.x * blockDim.x + threadIdx.x;
    int e    = gtid >> 5;
    int lane = gtid & 31;
    if (e >= E) return;
    int s = (int)src[e];
    int d = (int)dst[e];
    float norm = dinv[s] * dinv[d];
    float v = h[(size_t)s * HID + lane] * norm;
    atomicAdd(&agg[(size_t)d * HID + lane], v);
}

// ---------------------------------------------------------------------------
// Final MLP per node: out = relu(relu(x3) @ lin1_W + lin1_b) @ lin2_W + lin2_b
// ---------------------------------------------------------------------------
__global__ void mlp_kernel(const float* __restrict__ x3,
                           const float* __restrict__ lin1W,  // [32,16]
                           const float* __restrict__ lin1b,  // [16]
                           const float* __restrict__ lin2W,  // [16,10]
                           const float* __restrict__ lin2b,  // [10]
                           float* __restrict__ out, int n) {
    int i = blockIdx.x * blockDim.x + threadIdx.x;
    if (i >= n) return;

    float xr[32];
#pragma unroll
    for (int f = 0; f < 32; ++f) xr[f] = fmaxf(x3[(size_t)i * 32 + f], 0.0f);

    float t[16];
#pragma unroll
    for (int j = 0; j < 16; ++j) {
        float acc = lin1b[j];
#pragma unroll
        for (int f = 0; f < 32; ++f) acc = fmaf(xr[f], lin1W[f * 16 + j], acc);
        t[j] = fmaxf(acc, 0.0f);
    }
#pragma unroll
    for (int c = 0; c < 10; ++c) {
        float acc = lin2b[c];
#pragma unroll
        for (int j = 0; j < 16; ++j) acc = fmaf(t[j], lin2W[j * 10 + c], acc);
        out[(size_t)i * 10 + c] = acc;
    }
}

// ---------------------------------------------------------------------------
// Host-side orchestration
// ---------------------------------------------------------------------------
extern "C" void kernel_launch(void* const* d_in, const int* in_sizes, int n_in,
                              void* d_out, int out_size, void* d_ws, size_t ws_size,
                              hipStream_t stream) {
    const float*     x0    = (const float*)d_in[0];
    const long long* eidx  = (const long long*)d_in[1];
    // d_in[2] = batch (unused)
    const float* W1    = (const float*)d_in[3];
    const float* b1    = (const float*)d_in[4];
    const float* W2    = (const float*)d_in[5];
    const float* b2    = (const float*)d_in[6];
    const float* W3    = (const float*)d_in[7];
    const float* b3    = (const float*)d_in[8];
    const float* lin1W = (const float*)d_in[9];
    const float* lin1b = (const float*)d_in[10];
    const float* lin2W = (const float*)d_in[11];
    const float* lin2b = (const float*)d_in[12];

    const int N = in_sizes[0] / IN_FEAT;   // 100000
    const int E = in_sizes[1] / 2;         // 3200000
    const long long* src = eidx;
    const long long* dst = eidx + E;

    // workspace layout
    float* dinv = (float*)d_ws;                      // N floats (deg -> dinv)
    size_t dinvCnt = ((size_t)N + 127) & ~(size_t)127;
    float* h    = dinv + dinvCnt;                    // N*32 floats
    float* agg  = h + (size_t)N * HID;               // N*32 floats

    const int ntiles = N / 16;                       // 6250 (N divisible by 16)
    const int gemmBlocks = (ntiles + 7) / 8;
    const int nf = N * HID;

    // 1) degrees -> dinv
    zero_f32<<<(N + 255) / 256, 256, 0, stream>>>(dinv, N);
    degree_kernel<<<(E + 255) / 256, 256, 0, stream>>>(dst, dinv, E);
    rsqrt_kernel<<<(N + 255) / 256, 256, 0, stream>>>(dinv, N);

    // 2) layer 1: h = x0 @ W1 ; agg = h*dinv^2 + b1 ; scatter edges
    gemm_wmma<0><<<gemmBlocks, 256, IN_FEAT * HID * sizeof(float), stream>>>(
        x0, W1, h, ntiles, IN_FEAT);
    init_agg_kernel<<<(nf + 255) / 256, 256, 0, stream>>>(h, dinv, b1, agg, nf);
    scatter_kernel<<<(E * 32 + 255) / 256, 256, 0, stream>>>(src, dst, dinv, h, agg, E);

    // 3) layer 2: h = relu(agg) @ W2 ; agg = h*dinv^2 + b2 ; scatter
    gemm_wmma<1><<<gemmBlocks, 256, HID * HID * sizeof(float), stream>>>(
        agg, W2, h, ntiles, HID);
    init_agg_kernel<<<(nf + 255) / 256, 256, 0, stream>>>(h, dinv, b2, agg, nf);
    scatter_kernel<<<(E * 32 + 255) / 256, 256, 0, stream>>>(src, dst, dinv, h, agg, E);

    // 4) layer 3
    gemm_wmma<1><<<gemmBlocks, 256, HID * HID * sizeof(float), stream>>>(
        agg, W3, h, ntiles, HID);
    init_agg_kernel<<<(nf + 255) / 256, 256, 0, stream>>>(h, dinv, b3, agg, nf);
    scatter_kernel<<<(E * 32 + 255) / 256, 256, 0, stream>>>(src, dst, dinv, h, agg, E);

    // 5) final MLP (relu fused on load)
    mlp_kernel<<<(N + 255) / 256, 256, 0, stream>>>(agg, lin1W, lin1b, lin2W, lin2b,
                                                    (float*)d_out, N);
}